// RGAT_18777597018722
// MI455X (gfx1250) — compile-verified
//
#include <hip/hip_runtime.h>

#define N_NODES 50000
#define N_EDGES 800000
#define IN_DIM 128
#define OUT_DIM 64
#define NEG_SLOPE 0.2f
#define TOT_EDGES (N_EDGES + N_NODES)
#define N_STRIPES (N_NODES / 16)   // 3125, exact

typedef __attribute__((ext_vector_type(2))) float v2f;
typedef __attribute__((ext_vector_type(8))) float v8f;

// ---------------- order-preserving float <-> uint for atomic max ----------
__device__ __forceinline__ unsigned fflip(float f) {
    unsigned u = __float_as_uint(f);
    return (u & 0x80000000u) ? ~u : (u | 0x80000000u);
}
__device__ __forceinline__ float funflip(unsigned u) {
    u = (u & 0x80000000u) ? (u & 0x7FFFFFFFu) : ~u;
    return __uint_as_float(u);
}

// ---------------- 0: init out = bias, m_flip = 0 (-inf), denom = 0 -------
__global__ void gat_init(const float* __restrict__ bias, float* __restrict__ out,
                         unsigned* __restrict__ m_flip, float* __restrict__ denom) {
    int t = blockIdx.x * blockDim.x + threadIdx.x;
    if (t < N_NODES * OUT_DIM) out[t] = bias[t & (OUT_DIM - 1)];
    if (t < N_NODES) { m_flip[t] = 0u; denom[t] = 0.0f; }
}

// ---------------- 1: h = x @ W via V_WMMA_F32_16X16X4_F32 -----------------
// One wave per 16-row stripe, 4 accumulators covering all 64 output cols.
// B fragments come from a K-paired LDS layout: lpair[k/2][n] = {W[k][n], W[k+1][n]}
// so each fragment is a single aligned ds_load_b64 into an adjacent VGPR pair.
__global__ __launch_bounds__(256) void gat_gemm_wmma(const float* __restrict__ x,
                                                     const float* __restrict__ W,
                                                     float* __restrict__ h) {
    __shared__ float lrow[IN_DIM * OUT_DIM];              // 32 KB row-major staging
    __shared__ v2f   lpair[(IN_DIM / 2) * OUT_DIM];       // 32 KB K-paired layout
    const int tid = threadIdx.x;

    // Phase 1: coalesced global -> LDS (2048 float4 / 256 threads = 8 each)
    {
        const float4* Wv = (const float4*)W;
        float4* lv = (float4*)lrow;
        #pragma unroll
        for (int i = 0; i < (IN_DIM * OUT_DIM / 4) / 256; ++i)
            lv[tid + i * 256] = Wv[tid + i * 256];
    }
    __syncthreads();

    // Phase 2: repack to K-pairs (4096 elements / 256 threads = 16 each)
    {
        #pragma unroll
        for (int i = 0; i < (IN_DIM / 2) * OUT_DIM / 256; ++i) {
            int idx = tid + i * 256;
            int kp = idx >> 6;          // k/2
            int n  = idx & 63;
            v2f p;
            p.x = lrow[(2 * kp + 0) * OUT_DIM + n];
            p.y = lrow[(2 * kp + 1) * OUT_DIM + n];
            lpair[idx] = p;
        }
    }
    __syncthreads();

    const int wave   = tid >> 5;
    const int lane   = tid & 31;
    const int stripe = blockIdx.x * 8 + wave;
    if (stripe >= N_STRIPES) return;          // wave-uniform: EXEC stays all-ones
    const int row0 = stripe * 16;
    const int half = lane >> 4;               // 0: lanes 0-15, 1: lanes 16-31
    const int l16  = lane & 15;

    v8f c0 = {}, c1 = {}, c2 = {}, c3 = {};
    // A fragment: row = row0 + l16, K pair base = k0 + 2*half (ISA 16x4 f32 A layout)
    const float* xrow = x + (size_t)(row0 + l16) * IN_DIM + 2 * half;

    #pragma unroll 8
    for (int k0 = 0; k0 < IN_DIM; k0 += 4) {
        v2f a = *(const v2f*)(xrow + k0);             // contiguous global_load_b64
        const v2f* bp = lpair + ((k0 >> 1) + half) * OUT_DIM + l16;
        v2f b0 = bp[0];                               // single ds_load_b64 each
        v2f b1 = bp[16];
        v2f b2 = bp[32];
        v2f b3 = bp[48];
        c0 = __builtin_amdgcn_wmma_f32_16x16x4_f32(false, a, false, b0, (short)0, c0, false, false);
        c1 = __builtin_amdgcn_wmma_f32_16x16x4_f32(false, a, false, b1, (short)0, c1, false, false);
        c2 = __builtin_amdgcn_wmma_f32_16x16x4_f32(false, a, false, b2, (short)0, c2, false, false);
        c3 = __builtin_amdgcn_wmma_f32_16x16x4_f32(false, a, false, b3, (short)0, c3, false, false);
    }

    // C/D layout: VGPR r -> row r + 8*half, col = l16 + 16*ntile
    #pragma unroll
    for (int r = 0; r < 8; ++r) {
        float* o = h + (size_t)(row0 + r + 8 * half) * OUT_DIM + l16;
        o[0]  = c0[r];
        o[16] = c1[r];
        o[32] = c2[r];
        o[48] = c3[r];
    }
}

// ---------------- 2: per-node attention logits ----------------------------
__global__ void gat_logits(const float* __restrict__ h, const float* __restrict__ att_s,
                           const float* __restrict__ att_d, float* __restrict__ a_s,
                           float* __restrict__ a_d) {
    int i = blockIdx.x * blockDim.x + threadIdx.x;
    if (i >= N_NODES) return;
    const float* hi = h + (size_t)i * OUT_DIM;
    float s = 0.0f, d = 0.0f;
    #pragma unroll
    for (int n = 0; n < OUT_DIM; ++n) {
        float v = hi[n];
        s = fmaf(v, att_s[n], s);
        d = fmaf(v, att_d[n], d);
    }
    a_s[i] = s;
    a_d[i] = d;
}

// ---------------- 3: edge leaky-relu logits + segment max -----------------
__global__ void gat_edge_max(const int* __restrict__ ei, const float* __restrict__ a_s,
                             const float* __restrict__ a_d, float* __restrict__ e_arr,
                             unsigned* __restrict__ m_flip) {
    int t = blockIdx.x * blockDim.x + threadIdx.x;
    if (t >= TOT_EDGES) return;
    int s, d;
    if (t < N_EDGES) { s = ei[t]; d = ei[N_EDGES + t]; }
    else             { s = t - N_EDGES; d = s; }          // self loop
    float e = a_s[s] + a_d[d];
    e = (e > 0.0f) ? e : NEG_SLOPE * e;
    e_arr[t] = e;
    atomicMax(m_flip + d, fflip(e));
}

// ---------------- 4: exp(e - m) + segment sum -----------------------------
__global__ void gat_edge_expsum(const int* __restrict__ ei, float* __restrict__ e_arr,
                                const unsigned* __restrict__ m_flip,
                                float* __restrict__ denom) {
    int t = blockIdx.x * blockDim.x + threadIdx.x;
    if (t >= TOT_EDGES) return;
    int d = (t < N_EDGES) ? ei[N_EDGES + t] : (t - N_EDGES);
    float m  = funflip(m_flip[d]);
    float ex = __expf(e_arr[t] - m);
    e_arr[t] = ex;
    atomicAdd(denom + d, ex);
}

// ---------------- 5: weighted gather-scatter aggregation ------------------
// One wave per edge; lane handles output dims {lane, lane+32}.
__global__ void gat_aggregate(const int* __restrict__ ei, const float* __restrict__ h,
                              const float* __restrict__ ex_arr,
                              const float* __restrict__ denom, float* __restrict__ out) {
    long t = (long)blockIdx.x * blockDim.x + threadIdx.x;
    long edge = t >> 5;
    int lane  = (int)(t & 31);
    if (edge >= TOT_EDGES) return;
    int s, d;
    if (edge < N_EDGES) { s = ei[edge]; d = ei[N_EDGES + edge]; }
    else                { s = (int)edge - N_EDGES; d = s; }
    float alpha = ex_arr[edge] / denom[d];
    const float* hs = h + (size_t)s * OUT_DIM;
    float* od = out + (size_t)d * OUT_DIM;
    atomicAdd(od + lane,      hs[lane]      * alpha);
    atomicAdd(od + lane + 32, hs[lane + 32] * alpha);
}

extern "C" void kernel_launch(void* const* d_in, const int* in_sizes, int n_in,
                              void* d_out, int out_size, void* d_ws, size_t ws_size,
                              hipStream_t stream) {
    const float* x       = (const float*)d_in[0];
    const int*   ei      = (const int*)  d_in[1];   // [2, E] flat: row0=src, row1=dst
    const float* W       = (const float*)d_in[2];
    const float* att_src = (const float*)d_in[3];
    const float* att_dst = (const float*)d_in[4];
    const float* bias    = (const float*)d_in[5];
    float* out = (float*)d_out;
    (void)in_sizes; (void)n_in; (void)out_size; (void)ws_size;

    char* ws = (char*)d_ws;
    size_t off = 0;
    auto salloc = [&](size_t bytes) -> void* {
        void* p = ws + off;
        off = (off + bytes + 255) & ~(size_t)255;
        return p;
    };
    float*    h      = (float*)   salloc((size_t)N_NODES * OUT_DIM * sizeof(float));
    float*    a_src  = (float*)   salloc((size_t)N_NODES * sizeof(float));
    float*    a_dst  = (float*)   salloc((size_t)N_NODES * sizeof(float));
    unsigned* m_flip = (unsigned*)salloc((size_t)N_NODES * sizeof(unsigned));
    float*    denom  = (float*)   salloc((size_t)N_NODES * sizeof(float));
    float*    e_arr  = (float*)   salloc((size_t)TOT_EDGES * sizeof(float));

    // 0: out = bias, segment-max/-sum accumulators reset (every call; harness poisons once)
    gat_init<<<(N_NODES * OUT_DIM + 255) / 256, 256, 0, stream>>>(bias, out, m_flip, denom);
    // 1: h = x @ W (WMMA f32)
    gat_gemm_wmma<<<(N_STRIPES + 7) / 8, 256, 0, stream>>>(x, W, h);
    // 2: attention logits
    gat_logits<<<(N_NODES + 255) / 256, 256, 0, stream>>>(h, att_src, att_dst, a_src, a_dst);
    // 3: edge logits + segment max
    gat_edge_max<<<(TOT_EDGES + 255) / 256, 256, 0, stream>>>(ei, a_src, a_dst, e_arr, m_flip);
    // 4: exp + segment sum
    gat_edge_expsum<<<(TOT_EDGES + 255) / 256, 256, 0, stream>>>(ei, e_arr, m_flip, denom);
    // 5: weighted aggregation (one wave per edge)
    long agg_threads = (long)TOT_EDGES * 32;
    gat_aggregate<<<(int)((agg_threads + 255) / 256), 256, 0, stream>>>(ei, h, e_arr, denom, out);
}